// VNL_Loss_60722247631097
// MI455X (gfx1250) — compile-verified
//
#include <hip/hip_runtime.h>
#include <hip/hip_bf16.h>

// ---------------- problem constants (from reference) ----------------
#define IMG_H 480
#define IMG_W 640
#define FXC   519.0f
#define FYC   519.0f
#define U0C   320.0f
#define V0C   240.0f
#define DELTA_COS  0.867f
#define DELTA_DIFF 0.005f
#define DELTA_Z    0.0001f

#define SUM_NBLK 720

typedef __attribute__((ext_vector_type(2))) float v2f;
typedef __attribute__((ext_vector_type(8))) float v8f;

struct Meta {
    unsigned n_valid;
    unsigned kval;
    unsigned prefix;
    unsigned k_rem;
    unsigned hist[256];
};

// ---------------- CDNA5 WMMA wave32 reduction -----------------------
// A(16x4 f32) layout: VGPR0 = K=0 (lanes 0-15, M=lane) / K=2 (lanes 16-31,
// M=lane-16); VGPR1 = K=1 / K=3.  With A[m][0]=x_m, A[m][2]=x_{m+16},
// A[m][1]=A[m][3]=0 and B = ones(4x16):  D[m][n] = x_m + x_{m+16}.
// Lane n<16 holds D rows 0..7 in its 8 dest VGPRs, lanes>=16 rows 8..15,
// so summing the 8 components gives half-sums; one xor-16 shuffle finishes.
__device__ __forceinline__ float wave_reduce_wmma(float x) {
    v2f a; a[0] = x;    a[1] = 0.0f;
    v2f b; b[0] = 1.0f; b[1] = 1.0f;
    v8f c = {};
    v8f d = __builtin_amdgcn_wmma_f32_16x16x4_f32(
        false, a, false, b, (short)0, c, false, false);
    float s = d[0] + d[1] + d[2] + d[3] + d[4] + d[5] + d[6] + d[7];
    s += __shfl_xor(s, 16, 32);
    return s;
}

// ---------------- helpers -------------------------------------------
struct F3 { float x, y, z; };

__device__ __forceinline__ F3 back_project(const float* depth, int b, int p) {
    float dz = depth[b * (IMG_H * IMG_W) + p];
    int row = p / IMG_W;
    int col = p - row * IMG_W;
    float ad = fabsf(dz);
    F3 r;
    r.x = ((float)col - U0C) * ad / FXC;
    r.y = ((float)row - V0C) * ad / FYC;
    r.z = dz;
    return r;
}

__device__ __forceinline__ F3 sub3(F3 a, F3 b) {
    F3 r; r.x = a.x - b.x; r.y = a.y - b.y; r.z = a.z - b.z; return r;
}
__device__ __forceinline__ float dot3(F3 a, F3 b) {
    return a.x * b.x + a.y * b.y + a.z * b.z;
}
__device__ __forceinline__ F3 cross3(F3 a, F3 b) {
    F3 r;
    r.x = a.y * b.z - a.z * b.y;
    r.y = a.z * b.x - a.x * b.z;
    r.z = a.x * b.y - a.y * b.x;
    return r;
}
__device__ __forceinline__ float safe_norm3(F3 a) {
    float sq = dot3(a, a);
    return (sq == 0.0f) ? 0.0f : sqrtf(sq);
}

// ---------------- kernel 1: per-group lval + validity ----------------
__global__ void vnl_compute_kernel(const float* __restrict__ gt,
                                   const float* __restrict__ pr,
                                   const int* __restrict__ p1,
                                   const int* __restrict__ p2,
                                   const int* __restrict__ p3,
                                   unsigned* __restrict__ keys,
                                   Meta* __restrict__ meta,
                                   int G, int n_total) {
    int t = blockIdx.x * blockDim.x + threadIdx.x;
    if (t >= n_total) return;
    int b = t / G;
    int g = t - b * G;

    int P1 = p1[g], P2 = p2[g], P3 = p3[g];

    // gt 3D points
    F3 A = back_project(gt, b, P1);
    F3 Bp = back_project(gt, b, P2);
    F3 C = back_project(gt, b, P3);

    F3 d12 = sub3(Bp, A);
    F3 d13 = sub3(C, A);
    F3 d23 = sub3(C, Bp);

    float q0 = safe_norm3(d12);
    float q1 = safe_norm3(d13);
    float q2 = safe_norm3(d23);

    // Gram / cosine mask: count |ne_ij| > DELTA_COS over full 3x3 (symmetric)
    float e00 = dot3(d12, d12), e11 = dot3(d13, d13), e22 = dot3(d23, d23);
    float e01 = dot3(d12, d13), e02 = dot3(d12, d23), e12 = dot3(d13, d23);
    float ne00 = e00 / (q0 * q0 + 1e-8f);
    float ne11 = e11 / (q1 * q1 + 1e-8f);
    float ne22 = e22 / (q2 * q2 + 1e-8f);
    float ne01 = e01 / (q0 * q1 + 1e-8f);
    float ne02 = e02 / (q0 * q2 + 1e-8f);
    float ne12 = e12 / (q1 * q2 + 1e-8f);
    int cnt = (fabsf(ne00) > DELTA_COS) + (fabsf(ne11) > DELTA_COS) +
              (fabsf(ne22) > DELTA_COS) +
              2 * (fabsf(ne01) > DELTA_COS) + 2 * (fabsf(ne02) > DELTA_COS) +
              2 * (fabsf(ne12) > DELTA_COS);
    bool mask_cos = cnt > 3;

    bool mask_pad = (A.z > DELTA_Z) && (Bp.z > DELTA_Z) && (C.z > DELTA_Z);

    bool mask_x = (fabsf(d12.x) < DELTA_DIFF) || (fabsf(d13.x) < DELTA_DIFF) ||
                  (fabsf(d23.x) < DELTA_DIFF);
    bool mask_y = (fabsf(d12.y) < DELTA_DIFF) || (fabsf(d13.y) < DELTA_DIFF) ||
                  (fabsf(d23.y) < DELTA_DIFF);
    bool mask_z = (fabsf(d12.z) < DELTA_DIFF) || (fabsf(d13.z) < DELTA_DIFF) ||
                  (fabsf(d23.z) < DELTA_DIFF);

    bool mask = mask_pad && !((mask_x && mask_y && mask_z) || mask_cos);

    // pred 3D points
    F3 PA = back_project(pr, b, P1);
    F3 PB = back_project(pr, b, P2);
    F3 PC = back_project(pr, b, P3);

    // Reference broadcasting quirk: zmask over POINT index applied along the
    // COORDINATE axis -> z(point j)==0 sets coordinate j of all 3 points.
    bool zm0 = (PA.z == 0.0f);
    bool zm1 = (PB.z == 0.0f);
    bool zm2 = (PC.z == 0.0f);
    if (zm0) { PA.x = 0.0001f; PB.x = 0.0001f; PC.x = 0.0001f; }
    if (zm1) { PA.y = 0.0001f; PB.y = 0.0001f; PC.y = 0.0001f; }
    if (zm2) { PA.z = 0.0001f; PB.z = 0.0001f; PC.z = 0.0001f; }

    F3 gt_n = cross3(d12, d13);
    F3 dt_n = cross3(sub3(PB, PA), sub3(PC, PA));

    float gn = safe_norm3(gt_n); if (gn == 0.0f) gn = 0.01f;
    float dn = safe_norm3(dt_n); if (dn == 0.0f) dn = 0.01f;

    float lval = fabsf(gt_n.x / gn - dt_n.x / dn) +
                 fabsf(gt_n.y / gn - dt_n.y / dn) +
                 fabsf(gt_n.z / gn - dt_n.z / dn);

    unsigned key = 0u;
    if (mask) {
        key = __float_as_uint(lval) + 1u;   // lval >= 0 -> monotone encoding
        atomicAdd(&meta->n_valid, 1u);
    }
    keys[t] = key;
}

// ---------------- init ----------------------------------------------
__global__ void vnl_init_kernel(Meta* meta) {
    meta->hist[threadIdx.x] = 0u;
    if (threadIdx.x == 0) {
        meta->n_valid = 0u;
        meta->kval = 0u;
        meta->prefix = 0u;
        meta->k_rem = 0u;
    }
}

// ---------------- radix-select histogram ------------------------------
__global__ void vnl_hist_kernel(const unsigned* __restrict__ keys,
                                Meta* __restrict__ meta,
                                int pass, int n) {
    __shared__ unsigned lh[256];
    lh[threadIdx.x] = 0u;
    __syncthreads();
    unsigned prefix = meta->prefix;
    int stride = gridDim.x * blockDim.x;
    for (int i = blockIdx.x * blockDim.x + threadIdx.x; i < n; i += stride) {
        unsigned key = keys[i];
        if (pass == 0) {
            atomicAdd(&lh[key >> 24], 1u);
        } else {
            int sp = 32 - 8 * pass;
            if ((key >> sp) == prefix) {
                unsigned bkt = (key >> (24 - 8 * pass)) & 255u;
                atomicAdd(&lh[bkt], 1u);
            }
        }
    }
    __syncthreads();
    unsigned v = lh[threadIdx.x];
    if (v) atomicAdd(&meta->hist[threadIdx.x], v);
}

// ---------------- radix-select scan (serial over 256 bins) -----------
__global__ void vnl_scan_kernel(Meta* meta, int pass) {
    if (pass == 0) {
        unsigned nv = meta->n_valid;
        unsigned k = nv - nv / 4u;
        meta->kval = k;
        meta->k_rem = k;
    }
    if (meta->kval != 0u) {
        unsigned krem = meta->k_rem;
        unsigned cum = 0u;
        int sel = 0;
        for (int bin = 255; bin >= 0; --bin) {
            unsigned c = meta->hist[bin];
            if (cum + c >= krem) { sel = bin; break; }
            cum += c;
        }
        meta->prefix = (meta->prefix << 8) | (unsigned)sel;
        meta->k_rem = krem - cum;
    }
    for (int i = 0; i < 256; ++i) meta->hist[i] = 0u;
}

// ---------------- sum of values strictly above threshold --------------
__global__ void __launch_bounds__(256)
vnl_sum_kernel(const unsigned* __restrict__ keys,
               const Meta* __restrict__ meta,
               double* __restrict__ blockSums, int n) {
    __shared__ float wsum[8];
    unsigned k = meta->kval;
    unsigned T = meta->prefix;
    float local = 0.0f;
    if (k != 0u) {
        int chunk = (n + (int)gridDim.x - 1) / (int)gridDim.x;
        int start = (int)blockIdx.x * chunk;
        int end = start + chunk; if (end > n) end = n;
        for (int i = start + (int)threadIdx.x; i < end; i += (int)blockDim.x) {
            unsigned key = keys[i];
            if (key > T) local += __uint_as_float(key - 1u);
        }
    }
    // all 256 threads reach here together -> EXEC all ones for WMMA
    float tot = wave_reduce_wmma(local);
    int lane = threadIdx.x & 31;
    int wid = threadIdx.x >> 5;
    if (lane == 0) wsum[wid] = tot;
    __syncthreads();
    if (threadIdx.x == 0) {
        double s = 0.0;
        for (int w = 0; w < 8; ++w) s += (double)wsum[w];
        blockSums[blockIdx.x] = s;
    }
}

// ---------------- final: ordered sum + tie correction -----------------
__global__ void vnl_final_kernel(const double* __restrict__ blockSums,
                                 const Meta* __restrict__ meta,
                                 float* __restrict__ out, int nblk) {
    unsigned k = meta->kval;
    if (k == 0u) { out[0] = 0.0f; return; }
    double s = 0.0;
    for (int i = 0; i < nblk; ++i) s += blockSums[i];
    float tval = __uint_as_float(meta->prefix - 1u);
    double res = (s + (double)meta->k_rem * (double)tval) / (double)k;
    out[0] = (float)res;
}

// ---------------- launch ---------------------------------------------
extern "C" void kernel_launch(void* const* d_in, const int* in_sizes, int n_in,
                              void* d_out, int out_size, void* d_ws, size_t ws_size,
                              hipStream_t stream) {
    const float* gt = (const float*)d_in[0];
    const float* pr = (const float*)d_in[1];
    const int* p1 = (const int*)d_in[2];
    const int* p2 = (const int*)d_in[3];
    const int* p3 = (const int*)d_in[4];
    float* out = (float*)d_out;

    int G = in_sizes[2];
    int Bn = in_sizes[0] / (IMG_H * IMG_W);
    int n_total = Bn * G;

    char* ws = (char*)d_ws;
    double* blockSums = (double*)ws;                 // SUM_NBLK doubles
    Meta* meta = (Meta*)(ws + 8192);                 // counters + hist
    unsigned* keys = (unsigned*)(ws + 16384);        // n_total keys

    vnl_init_kernel<<<1, 256, 0, stream>>>(meta);

    int cblocks = (n_total + 255) / 256;
    vnl_compute_kernel<<<cblocks, 256, 0, stream>>>(gt, pr, p1, p2, p3,
                                                    keys, meta, G, n_total);

    int hblocks = cblocks < 1440 ? cblocks : 1440;
    for (int pass = 0; pass < 4; ++pass) {
        vnl_hist_kernel<<<hblocks, 256, 0, stream>>>(keys, meta, pass, n_total);
        vnl_scan_kernel<<<1, 1, 0, stream>>>(meta, pass);
    }

    vnl_sum_kernel<<<SUM_NBLK, 256, 0, stream>>>(keys, meta, blockSums, n_total);
    vnl_final_kernel<<<1, 1, 0, stream>>>(blockSums, meta, out, SUM_NBLK);
}